// CenterNeighAtt_18734647345347
// MI455X (gfx1250) — compile-verified
//
#include <hip/hip_runtime.h>
#include <stdint.h>
#include <math.h>

#define E 512
#define F 256
#define LEAKY 0.2f

typedef __attribute__((address_space(1))) int gint_t;
typedef __attribute__((address_space(3))) int lint_t;
typedef float v2f __attribute__((ext_vector_type(2)));
typedef float v8f __attribute__((ext_vector_type(8)));

// ---- CDNA5 async global->LDS copy (ASYNCcnt-tracked), with safe fallback ----
__device__ __forceinline__ void async_copy_f32(const float* g, float* l) {
#if __has_builtin(__builtin_amdgcn_global_load_async_to_lds_b32)
  __builtin_amdgcn_global_load_async_to_lds_b32((gint_t*)(uintptr_t)g,
                                                (lint_t*)(uintptr_t)l, 0, 0);
#else
  *l = *g;
#endif
}

__device__ __forceinline__ void wait_async0() {
#if __has_builtin(__builtin_amdgcn_s_wait_asynccnt)
  __builtin_amdgcn_s_wait_asynccnt(0);
#else
  asm volatile("s_wait_asynccnt 0" ::: "memory");
#endif
}

__device__ __forceinline__ float leaky(float x) {
  return x > 0.f ? x : LEAKY * x;
}

// ---------------------------------------------------------------------------
// Kernel A0: S = sum_r adj[r]   (elementwise, HBM-bound)
// ---------------------------------------------------------------------------
__global__ void cna_sum_r(const float* __restrict__ adj, float* __restrict__ S) {
  const size_t idx = ((size_t)blockIdx.x * 256 + threadIdx.x) * 4;
  const size_t P = (size_t)E * E;
  float4 a = *(const float4*)(adj + idx);
  float4 b = *(const float4*)(adj + P + idx);
  float4 c = *(const float4*)(adj + 2 * P + idx);
  float4 d = *(const float4*)(adj + 3 * P + idx);
  float4 r;
  r.x = (a.x + b.x) + (c.x + d.x);
  r.y = (a.y + b.y) + (c.y + d.y);
  r.z = (a.z + b.z) + (c.z + d.z);
  r.w = (a.w + b.w) + (c.w + d.w);
  *(float4*)(S + idx) = r;
}

// ---------------------------------------------------------------------------
// Kernel A1: in-place row softmax of S.
// One wave (32 lanes) per 16 rows.  Denominator rowsum = Exp(16xK) . ones(Kx16)
// accumulated with chained V_WMMA_F32_16X16X4_F32 (full fp32), row max via
// VALU + shfl_xor.  Streaming uses exactly the WMMA A-matrix layout:
//   lane m (m<16):  K=c0+0, c0+1   /   lane m+16:  K=c0+2, c0+3
// ---------------------------------------------------------------------------
__global__ void cna_row_softmax_wmma(float* __restrict__ S) {
  const int l    = threadIdx.x & 31;
  const int m    = l & 15;
  const int koff = (l >> 4) * 2;
  float* __restrict__ row = S + (size_t)(blockIdx.x * 16 + m) * E;

  // ---- pass 1: row max over this lane's column stream, combine halves ----
  float mxl = -INFINITY;
  for (int c0 = 0; c0 < E; c0 += 4) {
    const float2 x = *(const float2*)(row + c0 + koff);
    mxl = fmaxf(mxl, fmaxf(x.x, x.y));
  }
  const float mx = fmaxf(mxl, __shfl_xor(mxl, 16, 32));

  // ---- pass 2: rowsum of exp via WMMA (Exp . ones) ----
  float rowsum;
#if __has_builtin(__builtin_amdgcn_wmma_f32_16x16x4_f32)
  {
    v8f acc = {};
    const v2f ones = {1.f, 1.f};
    for (int c0 = 0; c0 < E; c0 += 4) {
      const float2 x = *(const float2*)(row + c0 + koff);
      v2f a;
      a.x = __expf(x.x - mx);
      a.y = __expf(x.y - mx);
      acc = __builtin_amdgcn_wmma_f32_16x16x4_f32(
          /*neg_a=*/false, a, /*neg_b=*/false, ones,
          /*c_mod=*/(short)0, acc, /*reuse_a=*/false, /*reuse_b=*/false);
    }
    // C/D layout: VGPR g holds M=g (lanes 0-15) and M=g+8 (lanes 16-31).
    // Lane q's acc[q&7] is the sum of row (q&7) + (q<16 ? 0 : 8).
    const int g = l & 7;
    const float own =
        (g & 4) ? ((g & 2) ? ((g & 1) ? acc[7] : acc[6])
                           : ((g & 1) ? acc[5] : acc[4]))
                : ((g & 2) ? ((g & 1) ? acc[3] : acc[2])
                           : ((g & 1) ? acc[1] : acc[0]));
    const int src = ((m & 8) << 1) | (m & 7);  // lane holding row m's sum
    rowsum = __shfl(own, src, 32);
  }
#else
  {
    float s = 0.f;
    for (int c0 = 0; c0 < E; c0 += 4) {
      const float2 x = *(const float2*)(row + c0 + koff);
      s += __expf(x.x - mx) + __expf(x.y - mx);
    }
    rowsum = s + __shfl_xor(s, 16, 32);
  }
#endif
  const float inv = 1.0f / rowsum;

  // ---- pass 3: normalize in place ----
  for (int c0 = 0; c0 < E; c0 += 4) {
    float2 x = *(const float2*)(row + c0 + koff);
    x.x = __expf(x.x - mx) * inv;
    x.y = __expf(x.y - mx) * inv;
    *(float2*)(row + c0 + koff) = x;
  }
}

// ---------------------------------------------------------------------------
// Kernel B: per (j,f): m = max_i att, invZ = 1/sum_i exp(att-m)
//           att[i] = leaky_relu( s[i,j] * (h[i,f]*h[j,f]) )
// One block per j (strided s column staged into LDS via async engine),
// one thread per f.  mz packs (m, invZ) as float2.
// ---------------------------------------------------------------------------
__global__ void cna_col_stats(const float* __restrict__ h,
                              const float* __restrict__ s,
                              float2* __restrict__ mz) {
  const int j = blockIdx.x;
  const int t = threadIdx.x;
  __shared__ float scol[E];

  async_copy_f32(&s[(size_t)t * E + j],         &scol[t]);
  async_copy_f32(&s[(size_t)(t + 256) * E + j], &scol[t + 256]);
  wait_async0();
  __syncthreads();

  const int f = t;
  const float hj = h[(size_t)j * F + f];

  float mx = -INFINITY;
  for (int i = 0; i < E; ++i) {
    const float x = leaky(scol[i] * (h[(size_t)i * F + f] * hj));
    mx = fmaxf(mx, x);
  }
  float sum = 0.f;
  for (int i = 0; i < E; ++i) {
    const float x = leaky(scol[i] * (h[(size_t)i * F + f] * hj));
    sum += __expf(x - mx);
  }
  mz[(size_t)j * F + f] = make_float2(mx, 1.0f / sum);
}

// ---------------------------------------------------------------------------
// Kernel C: h_prime[i,f] = sum_j exp(att - m[j,f]) * invZ[j,f] * h[j,f]
//           out = elu(h_prime);  alpha = 1/E (analytic collapse: softmax over
//           i sums to 1, so colsum(scores) is constant in j)
// ---------------------------------------------------------------------------
__global__ void cna_aggregate(const float* __restrict__ h,
                              const float* __restrict__ s,
                              const float2* __restrict__ mz,
                              float* __restrict__ out) {
  const int i = blockIdx.x;
  const int t = threadIdx.x;
  __shared__ float srow[E];

  async_copy_f32(&s[(size_t)i * E + t],       &srow[t]);
  async_copy_f32(&s[(size_t)i * E + 256 + t], &srow[t + 256]);
  wait_async0();
  __syncthreads();

  const int f = t;
  const float hi = h[(size_t)i * F + f];

  float acc = 0.f;
  for (int j = 0; j < E; ++j) {
    if (j + 16 < E) {
      __builtin_prefetch(&h[(size_t)(j + 16) * F + f], 0, 0);
      __builtin_prefetch(&mz[(size_t)(j + 16) * F + f], 0, 0);
    }
    const float  hj = h[(size_t)j * F + f];
    const float2 p  = mz[(size_t)j * F + f];
    const float  x  = leaky(srow[j] * (hi * hj));
    const float  a  = __expf(x - p.x) * p.y;
    acc = fmaf(a, hj, acc);
  }

  out[(size_t)i * F + f] = acc > 0.f ? acc : expm1f(acc);

  if (i == 0) {
    out[(size_t)E * F + f]       = 1.0f / E;
    out[(size_t)E * F + 256 + f] = 1.0f / E;
  }
}

// ---------------------------------------------------------------------------
extern "C" void kernel_launch(void* const* d_in, const int* in_sizes, int n_in,
                              void* d_out, int out_size, void* d_ws, size_t ws_size,
                              hipStream_t stream) {
  (void)in_sizes; (void)n_in; (void)out_size; (void)ws_size;
  const float* h   = (const float*)d_in[0];   // [E,F]
  const float* adj = (const float*)d_in[1];   // [R,E,E]
  // d_in[2] = lin_w, d_in[3] = lin_b : analytically dead (alpha is uniform)

  float*  ws = (float*)d_ws;
  float*  s  = ws;                              // E*E floats (raw sums, then softmax in place)
  float2* mz = (float2*)(ws + (size_t)E * E);   // E*F float2

  float* out = (float*)d_out;

  cna_sum_r           <<<256,   256, 0, stream>>>(adj, s);
  cna_row_softmax_wmma<<<E / 16, 32, 0, stream>>>(s);
  cna_col_stats       <<<E,     256, 0, stream>>>(h, s, mz);
  cna_aggregate       <<<E,     256, 0, stream>>>(h, s, mz, out);
}